// DotProduct_11982958756713
// MI455X (gfx1250) — compile-verified
//
#include <hip/hip_runtime.h>

typedef float v2f __attribute__((ext_vector_type(2)));
typedef float v8f __attribute__((ext_vector_type(8)));
typedef unsigned int u32x4 __attribute__((ext_vector_type(4)));
typedef unsigned int u32x8 __attribute__((ext_vector_type(8)));

#define N_Q   16384
#define M_C   16384
#define F_DIM 512
#define TILE  128
#define KC    32
#define LDSS  36   // KC + 4 pad (TDM pad_amount=4 DWORDs every 32 DWORDs)

// ---- Tensor Data Mover: DMA a TILE x KC fp32 tile (row stride F_DIM) into LDS,
// padding each 32-DWORD row with 4 DWORDs so the LDS stride is LDSS floats. ----
// D# layout per CDNA5 ISA.
static __device__ __forceinline__ void tdm_load_tile(unsigned lds_off,
                                                     const float* gptr) {
    const unsigned long long ga = (unsigned long long)gptr;

    u32x4 g0;
    g0.x = 1u;                                   // count=1, user descriptor
    g0.y = lds_off;                              // lds_addr (bytes)
    g0.z = (unsigned)ga;                         // global_addr[31:0]
    g0.w = ((unsigned)(ga >> 32) & 0x01FFFFFFu)  // global_addr[56:32]
         | (2u << 30);                           // type = 2 ("image")

    u32x8 g1;
    g1.s0 = (2u << 16)        // data_size = 4 bytes
          | (1u << 20)        // pad_enable
          | (4u << 22)        // pad_interval: code 4 -> every 32 DWORDs
          | (3u << 25);       // pad_amount:   code 3 -> 4 DWORDs
    g1.s1 = ((unsigned)F_DIM & 0xFFFFu) << 16;                 // tensor_dim0[15:0]
    g1.s2 = (((unsigned)F_DIM >> 16) & 0xFFFFu)                // tensor_dim0[31:16]
          | (((unsigned)N_Q & 0xFFFFu) << 16);                 // tensor_dim1[15:0]
    g1.s3 = (((unsigned)N_Q >> 16) & 0xFFFFu)                  // tensor_dim1[31:16]
          | ((unsigned)KC << 16);                              // tile_dim0 = 32
    g1.s4 = (unsigned)TILE;                                    // tile_dim1 = 128 (tile_dim2 = 0)
    g1.s5 = (unsigned)F_DIM;                                   // tensor_dim0_stride lo32
    g1.s6 = 0u;                                                // stride hi16, dim1_stride lo16
    g1.s7 = 0u;

    const u32x4 gz = {0u, 0u, 0u, 0u};                         // groups 2/3: unused dims

    asm volatile("tensor_load_to_lds %0, %1, %2, %3"
                 :
                 : "s"(g0), "s"(g1), "s"(gz), "s"(gz)
                 : "memory");
}

__global__ __launch_bounds__(256)
void batched_sim_wmma_kernel(const float* __restrict__ Hq,
                             const float* __restrict__ Hc,
                             const int*  __restrict__ batch_q,
                             const int*  __restrict__ batch_c,
                             float* __restrict__ out,
                             float* __restrict__ mask_out)
{
    __shared__ float As[2][TILE][LDSS];
    __shared__ float Bs[2][TILE][LDSS];
    __shared__ int   bq_s[TILE];
    __shared__ int   bc_s[TILE];

    const int tid  = threadIdx.x;
    const int lane = tid & 31;
    const int wid  = tid >> 5;        // 0..7
    const int half = lane >> 4;       // 0 or 1
    const int ln16 = lane & 15;

    const int rowBase = blockIdx.y * TILE;   // query rows
    const int colBase = blockIdx.x * TILE;   // candidate rows (output cols)

    // 8 waves arranged 4 (rows) x 2 (cols); each wave: 32 rows x 64 cols
    const int wRow = (wid >> 1) * 32;
    const int wCol = (wid & 1)  * 64;

    if (tid < TILE) {
        bq_s[tid] = batch_q[rowBase + tid];
        bc_s[tid] = batch_c[colBase + tid];
    }

    // ---- prologue: DMA first K-chunk of both tiles via TDM ----
    if (wid == 0)
        tdm_load_tile((unsigned)(uintptr_t)&As[0][0][0],
                      Hq + (size_t)rowBase * F_DIM);
    if (wid == 1)
        tdm_load_tile((unsigned)(uintptr_t)&Bs[0][0][0],
                      Hc + (size_t)colBase * F_DIM);
    if (wid < 2)
        __builtin_amdgcn_s_wait_tensorcnt(0);
    __syncthreads();

    v8f acc[2][4];
#pragma unroll
    for (int mi = 0; mi < 2; ++mi)
#pragma unroll
        for (int ni = 0; ni < 4; ++ni)
            acc[mi][ni] = (v8f){0.f, 0.f, 0.f, 0.f, 0.f, 0.f, 0.f, 0.f};

    int cur = 0;
    for (int k0 = 0; k0 < F_DIM; k0 += KC) {
        const int nxt = cur ^ 1;

        // ---- kick off DMA of the next chunk while we compute this one ----
        if (k0 + KC < F_DIM) {
            if (wid == 0)
                tdm_load_tile((unsigned)(uintptr_t)&As[nxt][0][0],
                              Hq + (size_t)rowBase * F_DIM + k0 + KC);
            if (wid == 1)
                tdm_load_tile((unsigned)(uintptr_t)&Bs[nxt][0][0],
                              Hc + (size_t)colBase * F_DIM + k0 + KC);
        }

        // ---- 8 x (2x4) WMMA f32 16x16x4 on the current chunk ----
#pragma unroll 4
        for (int kk = 0; kk < KC; kk += 4) {
            const int ka = kk + half * 2;  // lanes 0-15: K={kk,kk+1}; 16-31: K={kk+2,kk+3}
            const v2f a0 = *reinterpret_cast<const v2f*>(&As[cur][wRow +  0 + ln16][ka]);
            const v2f a1 = *reinterpret_cast<const v2f*>(&As[cur][wRow + 16 + ln16][ka]);
            const v2f b0 = *reinterpret_cast<const v2f*>(&Bs[cur][wCol +  0 + ln16][ka]);
            const v2f b1 = *reinterpret_cast<const v2f*>(&Bs[cur][wCol + 16 + ln16][ka]);
            const v2f b2 = *reinterpret_cast<const v2f*>(&Bs[cur][wCol + 32 + ln16][ka]);
            const v2f b3 = *reinterpret_cast<const v2f*>(&Bs[cur][wCol + 48 + ln16][ka]);

            acc[0][0] = __builtin_amdgcn_wmma_f32_16x16x4_f32(false, a0, false, b0, (short)0, acc[0][0], false, false);
            acc[0][1] = __builtin_amdgcn_wmma_f32_16x16x4_f32(false, a0, false, b1, (short)0, acc[0][1], false, false);
            acc[0][2] = __builtin_amdgcn_wmma_f32_16x16x4_f32(false, a0, false, b2, (short)0, acc[0][2], false, false);
            acc[0][3] = __builtin_amdgcn_wmma_f32_16x16x4_f32(false, a0, false, b3, (short)0, acc[0][3], false, false);
            acc[1][0] = __builtin_amdgcn_wmma_f32_16x16x4_f32(false, a1, false, b0, (short)0, acc[1][0], false, false);
            acc[1][1] = __builtin_amdgcn_wmma_f32_16x16x4_f32(false, a1, false, b1, (short)0, acc[1][1], false, false);
            acc[1][2] = __builtin_amdgcn_wmma_f32_16x16x4_f32(false, a1, false, b2, (short)0, acc[1][2], false, false);
            acc[1][3] = __builtin_amdgcn_wmma_f32_16x16x4_f32(false, a1, false, b3, (short)0, acc[1][3], false, false);
        }

        // issuing waves ensure their DMA landed; everyone syncs before swapping buffers
        if (wid < 2)
            __builtin_amdgcn_s_wait_tensorcnt(0);
        __syncthreads();
        cur = nxt;
    }

    // ---- epilogue: mask + store scores and mask (store-bandwidth bound) ----
#pragma unroll
    for (int mi = 0; mi < 2; ++mi) {
#pragma unroll
        for (int ni = 0; ni < 4; ++ni) {
            const int colL = wCol + ni * 16 + ln16;
            const int col  = colBase + colL;
            const int bc   = bc_s[colL];
#pragma unroll
            for (int v = 0; v < 8; ++v) {
                const int rowL = wRow + mi * 16 + v + 8 * half;
                const int row  = rowBase + rowL;
                const bool same = (bq_s[rowL] == bc);
                const size_t o  = (size_t)row * M_C + col;
                out[o]      = same ? acc[mi][ni][v] : -__builtin_inff();
                mask_out[o] = same ? 1.0f : 0.0f;
            }
        }
    }
}

extern "C" void kernel_launch(void* const* d_in, const int* in_sizes, int n_in,
                              void* d_out, int out_size, void* d_ws, size_t ws_size,
                              hipStream_t stream) {
    const float* Hq = (const float*)d_in[0];
    const float* Hc = (const float*)d_in[1];
    const int*   bq = (const int*)d_in[2];
    const int*   bc = (const int*)d_in[3];
    float* out   = (float*)d_out;
    float* maskp = out + (size_t)N_Q * M_C;   // tuple output: (out, mask) concatenated

    dim3 grid(M_C / TILE, N_Q / TILE);   // 128 x 128 workgroups
    batched_sim_wmma_kernel<<<grid, 256, 0, stream>>>(Hq, Hc, bq, bc, out, maskp);
}